// CFocalLoss_23055384445358
// MI455X (gfx1250) — compile-verified
//
#include <hip/hip_runtime.h>

typedef float v2f __attribute__((ext_vector_type(2)));
typedef float v8f __attribute__((ext_vector_type(8)));

#define WPB   8     // waves per block (wave32)
#define TPB   256   // threads per block
#define NITER 8     // 8 * 128 = 1024 columns max (C = 1000 fits; requires C % 4 == 0)

// wave32 xor-butterfly via ds_swizzle (group-of-32 xor mode): 1 DS op per step
// instead of the ~6-op generic __shfl_xor expansion.
#define SWZ_XOR(v, m) \
  __int_as_float(__builtin_amdgcn_ds_swizzle(__float_as_int(v), (((m) << 10) | 0x1F)))

__device__ __forceinline__ float waveReduceSum(float v) {
  v += SWZ_XOR(v, 16);
  v += SWZ_XOR(v, 8);
  v += SWZ_XOR(v, 4);
  v += SWZ_XOR(v, 2);
  v += SWZ_XOR(v, 1);
  return v;
}
__device__ __forceinline__ float waveReduceMax(float v) {
  v = fmaxf(v, SWZ_XOR(v, 16));
  v = fmaxf(v, SWZ_XOR(v, 8));
  v = fmaxf(v, SWZ_XOR(v, 4));
  v = fmaxf(v, SWZ_XOR(v, 2));
  v = fmaxf(v, SWZ_XOR(v, 1));
  return v;
}

// One wave per row: row kept register-resident (8 x float4 per lane),
// single pass over HBM (131 MB -> ~5.6 us at 23.3 TB/s is the floor).
// Main loop computes only the `neg` focal term (1 exp + 1 log per element);
// the lone target column is patched afterwards so the transcendental pipe
// never computes both logs per element.
__global__ __launch_bounds__(TPB) void focal_rows_kernel(
    const float* __restrict__ pred, const int* __restrict__ target,
    float* __restrict__ partial, int B, int C)
{
  const int lane = threadIdx.x & 31;
  const int wave = threadIdx.x >> 5;
  const int row  = blockIdx.x * WPB + wave;
  const int rowc = (row < B) ? row : 0;
  const float rowValid = (row < B) ? 1.0f : 0.0f;

  const float* __restrict__ rp = pred + (size_t)rowc * (size_t)C;
  const float NEGBIG = -3.0e38f;

  // ---- load row into registers (128 cols per wave iteration) ----
  float4 x[NITER];
  #pragma unroll
  for (int j = 0; j < NITER; ++j) {
    int col = j * 128 + lane * 4;
    if (col < C) {               // C % 4 == 0: vectors are never partial
      x[j] = *(const float4*)(rp + col);
    } else {
      x[j] = make_float4(NEGBIG, NEGBIG, NEGBIG, NEGBIG);
    }
  }

  // ---- phase 1: row max, then sum of exp(x - max) ----
  float m = NEGBIG;
  #pragma unroll
  for (int j = 0; j < NITER; ++j)
    m = fmaxf(m, fmaxf(fmaxf(x[j].x, x[j].y), fmaxf(x[j].z, x[j].w)));
  m = waveReduceMax(m);

  float s = 0.0f;
  #pragma unroll
  for (int j = 0; j < NITER; ++j) {
    // masked slots hold -3e38 -> expf underflows to exactly 0
    s += __expf(x[j].x - m);
    s += __expf(x[j].y - m);
    s += __expf(x[j].z - m);
    s += __expf(x[j].w - m);
  }
  s = waveReduceSum(s);
  const float inv = 1.0f / s;

  // ---- phase 2: neg term for every column; remember p at the target ----
  // Masked slots: p = 1e-9 -> term ~ -7e-28, far below fp32 resolution of
  // the result (-O(5) per row), so no per-element validity guard needed.
  const int tgt = target[rowc];
  float acc = 0.0f;
  float pt  = -1.0f;                       // sentinel: "target not in this lane"
  #pragma unroll
  for (int j = 0; j < NITER; ++j) {
    const int colBase = j * 128 + lane * 4;
    float vx[4] = { x[j].x, x[j].y, x[j].z, x[j].w };
    #pragma unroll
    for (int k = 0; k < 4; ++k) {
      float e = __expf(vx[k] - m);
      float p = e * inv + 1e-9f;                   // softmax + EPS
      acc += 0.5f * p * p * __log2f(1.0f - p);     // ALPHA * p^2 * log2(1-p)
      pt = (colBase + k == tgt) ? p : pt;          // 1 cmp + 1 cndmask
    }
  }

  // patch the single target column: remove its neg term, add its pos term
  if (pt >= 0.0f) {                                // true in exactly one lane
    float q = 1.0f - pt;
    acc += 0.5f * q * q * __log2f(pt)              // ALPHA*(1-p)^2*log2(p)
         - 0.5f * pt * pt * __log2f(q);
  }

  acc = waveReduceSum(acc) * rowValid;

  // ---- block partial (fixed-order => deterministic) ----
  __shared__ float wacc[WPB];
  if (lane == 0) wacc[wave] = acc;
  __syncthreads();
  if (threadIdx.x == 0) {
    float t = 0.0f;
    #pragma unroll
    for (int i = 0; i < WPB; ++i) t += wacc[i];
    partial[blockIdx.x] = t;
  }
}

// Final reduction: 256 threads fold n partials into 256 values, then wave 0
// sums them on the matrix pipe: D = A x ones + C  => every D column holds the
// row sums (16x replicated), so grand total = (sum of D)/16.
__global__ __launch_bounds__(TPB) void focal_reduce_kernel(
    const float* __restrict__ partial, float* __restrict__ out,
    int n, float scale)
{
  __shared__ float sums[TPB];
  const int t = threadIdx.x;
  float a = 0.0f;
  for (int i = t; i < n; i += TPB) a += partial[i];   // fixed stride order
  sums[t] = a;
  __syncthreads();

  if (t < 32) {
#if __has_builtin(__builtin_amdgcn_wmma_f32_16x16x4_f32)
    v8f c = {};
    v2f b = { 1.0f, 1.0f };
    #pragma unroll
    for (int it = 0; it < TPB / 64; ++it) {
      v2f av;
      av[0] = sums[it * 64 + t * 2 + 0];
      av[1] = sums[it * 64 + t * 2 + 1];
      // D = A(16x4 f32) x B(4x16 ones) + C  -> accumulate 64 values/step
      c = __builtin_amdgcn_wmma_f32_16x16x4_f32(
              false, av, false, b, (short)0, c, false, false);
    }
    float sTot = c[0] + c[1] + c[2] + c[3] + c[4] + c[5] + c[6] + c[7];
    sTot = waveReduceSum(sTot);
    if (t == 0) out[0] = sTot * 0.0625f * scale;  // /16: columns replicate sums
#else
    float sTot = 0.0f;
    #pragma unroll
    for (int i = 0; i < TPB / 32; ++i) sTot += sums[t + 32 * i];
    sTot = waveReduceSum(sTot);
    if (t == 0) out[0] = sTot * scale;
#endif
  }
}

extern "C" void kernel_launch(void* const* d_in, const int* in_sizes, int n_in,
                              void* d_out, int out_size, void* d_ws, size_t ws_size,
                              hipStream_t stream) {
  const float* pred   = (const float*)d_in[0];
  const int*   target = (const int*)d_in[1];
  float*       out    = (float*)d_out;
  float*       part   = (float*)d_ws;

  const int B = in_sizes[1];                 // 32768
  const int C = in_sizes[0] / in_sizes[1];   // 1000 (kernel supports C <= 1024, C % 4 == 0)
  const int blocks = (B + WPB - 1) / WPB;    // 4096 partials

  focal_rows_kernel<<<blocks, TPB, 0, stream>>>(pred, target, part, B, C);

  const float scale = (float)(-1.0 / ((double)B * (double)C));
  focal_reduce_kernel<<<1, TPB, 0, stream>>>(part, out, blocks, scale);
}